// HGNAM_13477607375494
// MI455X (gfx1250) — compile-verified
//
#include <hip/hip_runtime.h>
#include <hip/hip_bf16.h>

typedef __attribute__((ext_vector_type(2))) float v2f;
typedef __attribute__((ext_vector_type(8))) float v8f;

#define NN 2048
#define FF 64
#define HH 32
#define CC 16

// ============================================================================
// Kernel A: f_sums[n,c] = sum_k relu(x[n,k>>5]*W1[k]+b1[k]) * (attn[k>>5]*W2[k,c])
//                         + sum_f attn[f]*b2[f,c]
// GEMM M=2048, K=2048, N=16 via V_WMMA_F32_16X16X4_F32.
// 128 blocks x 256 thr (8 waves). Block -> 16 rows. Wave w -> K in [w*256, w*256+256).
// ============================================================================
__global__ __launch_bounds__(256)
void fsums_kernel(const float* __restrict__ x,    // [N,F]
                  const float* __restrict__ W1,   // [F,H] flat == [K]
                  const float* __restrict__ b1,   // [F,H] flat == [K]
                  const float* __restrict__ W2,   // [F,H,C] flat == [K,C]
                  const float* __restrict__ b2,   // [F,C]
                  const float* __restrict__ fw,   // [F]
                  float* __restrict__ fsums)      // [N,C]
{
    __shared__ float attnLds[FF];
    __shared__ float redC[8 * 256];

    const int tid  = threadIdx.x;
    const int lane = tid & 31;
    const int wave = tid >> 5;
    const int tileM = blockIdx.x * 16;

    // attn[f] = exp(exp(fw[f])) / sum  (fw in [0,1] -> values in [2.7,15.2], safe unstabilized)
    if (tid < FF) attnLds[tid] = __expf(__expf(fw[tid]));
    __syncthreads();
    float s = 0.f;
    #pragma unroll
    for (int f = 0; f < FF; ++f) s += attnLds[f];
    const float invS = 1.0f / s;

    const int row = lane & 15;     // M index within tile / N index for B
    const int hi  = lane >> 4;     // selects K pair {0,1} vs {2,3}
    const int col = row;
    const int gRow = tileM + row;

    v8f acc = {};
    const int kBase = wave * 256;
    for (int k0 = kBase; k0 < kBase + 256; k0 += 4) {
        const int kA = k0 + 2 * hi;        // this lane's two K indices: kA, kA+1
        const int f0 = kA >> 5;            // kA even -> same f for both
        // A fragment: relu(x*W1+b1), built directly in 16x4 f32 A layout
        const float xv  = x[gRow * FF + f0];
        v2f a;
        a.x = fmaxf(fmaf(xv, W1[kA],     b1[kA]),     0.f);
        a.y = fmaxf(fmaf(xv, W1[kA + 1], b1[kA + 1]), 0.f);
        // B fragment: attn-scaled W2, 4x16 f32 B layout
        const float at = attnLds[f0] * invS;
        v2f b;
        b.x = at * W2[kA       * CC + col];
        b.y = at * W2[(kA + 1) * CC + col];
        acc = __builtin_amdgcn_wmma_f32_16x16x4_f32(false, a, false, b,
                                                    (short)0, acc, false, false);
    }

    // cross-wave reduction of the 8 partial 16x16 tiles
    #pragma unroll
    for (int r = 0; r < 8; ++r) {
        const int orow = r + 8 * hi;
        redC[wave * 256 + orow * 16 + col] = acc[r];
    }
    __syncthreads();
    float v = 0.f;
    #pragma unroll
    for (int w = 0; w < 8; ++w) v += redC[w * 256 + tid];
    // bias term: sum_f attn[f]*b2[f,c]
    const int ocol = tid & 15;
    float bias = 0.f;
    #pragma unroll 4
    for (int f = 0; f < FF; ++f) bias += attnLds[f] * invS * b2[f * CC + ocol];
    fsums[tileM * CC + tid] = v + bias;   // (tileM + tid/16)*16 + tid%16 == tileM*16 + tid
}

// ============================================================================
// Kernel B: out = (m / norm) @ f_sums, m[n,k] = bm2 + sum_h Wm2[h]*relu(d*Wm1[h]+bm1[h])
// Fused: m never hits memory. GEMM M=2048, K=2048, N=16 via V_WMMA_F32_16X16X4_F32.
// 128 blocks x 256 thr. Block -> 16 rows. Wave w -> K chunk of 256, staged 64 at a time
// through a padded LDS tile (stride 66 dwords) to bridge coalesced global loads and
// the WMMA A layout. dist/norm streamed with NT hint (read exactly once),
// 1/norm via v_rcp_f32 instead of the full-precision divide expansion.
// ============================================================================
__global__ __launch_bounds__(256)
void out_kernel(const float* __restrict__ dist,   // [N,N]
                const float* __restrict__ norm,   // [N,N]
                const float* __restrict__ pWm1,   // [H]
                const float* __restrict__ pbm1,   // [H]
                const float* __restrict__ pWm2,   // [H]
                const float* __restrict__ pbm2,   // [1]
                const float* __restrict__ fsums,  // [N,C]
                float* __restrict__ out)          // [N,C]
{
    __shared__ float mT[8][16 * 66];   // per-wave m tile, padded rows (66 dwords)
    __shared__ float redC[8 * 256];

    const int tid  = threadIdx.x;
    const int lane = tid & 31;
    const int wave = tid >> 5;
    const int tileM = blockIdx.x * 16;
    const int row = lane & 15;
    const int hi  = lane >> 4;
    const int col = row;

    // wave-uniform m-network params (compiler scalarizes -> SGPR/K$)
    float wm1[HH], bb1[HH], wm2[HH];
    #pragma unroll
    for (int h = 0; h < HH; ++h) { wm1[h] = pWm1[h]; bb1[h] = pbm1[h]; wm2[h] = pWm2[h]; }
    const float bm2v = pbm2[0];

    float* myT = mT[wave];
    v8f acc = {};
    const int kBase = wave * 256;
    for (int kc = kBase; kc < kBase + 256; kc += 64) {
        // Stage: m for 16 rows x 64 cols into LDS. Loads are fully coalesced:
        // 32 lanes x float2 = 256B contiguous per row, nontemporal (single-use stream).
        for (int r = 0; r < 16; ++r) {
            const size_t base = (size_t)(tileM + r) * NN + kc + 2 * lane;
            const v2f d2 = __builtin_nontemporal_load((const v2f*)(dist + base));
            const v2f n2 = __builtin_nontemporal_load((const v2f*)(norm + base));
            float a0 = bm2v, a1 = bm2v;
            #pragma unroll
            for (int h = 0; h < HH; ++h) {
                const float t0 = fmaf(d2.x, wm1[h], bb1[h]);
                const float t1 = fmaf(d2.y, wm1[h], bb1[h]);
                a0 = fmaf(fmaxf(t0, 0.f), wm2[h], a0);
                a1 = fmaf(fmaxf(t1, 0.f), wm2[h], a1);
            }
            v2f mv;
            mv.x = a0 * __builtin_amdgcn_rcpf(n2.x);   // v_rcp_f32 + v_mul (vs ~8-op divide)
            mv.y = a1 * __builtin_amdgcn_rcpf(n2.y);
            *(v2f*)&myT[r * 66 + 2 * lane] = mv;       // conflict-free banks
        }
        // Same-wave LDS ops are in-order; compiler inserts the ds waits.
        #pragma unroll
        for (int kk = 0; kk < 64; kk += 4) {
            const v2f a = *(const v2f*)&myT[row * 66 + kk + 2 * hi];
            const int kg = kc + kk + 2 * hi;
            v2f b;
            b.x = fsums[kg       * CC + col];
            b.y = fsums[(kg + 1) * CC + col];
            acc = __builtin_amdgcn_wmma_f32_16x16x4_f32(false, a, false, b,
                                                        (short)0, acc, false, false);
        }
    }

    #pragma unroll
    for (int r = 0; r < 8; ++r) {
        const int orow = r + 8 * hi;
        redC[wave * 256 + orow * 16 + col] = acc[r];
    }
    __syncthreads();
    float v = 0.f;
    #pragma unroll
    for (int w = 0; w < 8; ++w) v += redC[w * 256 + tid];
    out[tileM * CC + tid] = v;
}

extern "C" void kernel_launch(void* const* d_in, const int* in_sizes, int n_in,
                              void* d_out, int out_size, void* d_ws, size_t ws_size,
                              hipStream_t stream) {
    const float* x    = (const float*)d_in[0];
    const float* dist = (const float*)d_in[1];
    const float* norm = (const float*)d_in[2];
    const float* W1   = (const float*)d_in[3];
    const float* b1   = (const float*)d_in[4];
    const float* W2   = (const float*)d_in[5];
    const float* b2   = (const float*)d_in[6];
    const float* fw   = (const float*)d_in[7];
    const float* Wm1  = (const float*)d_in[8];
    const float* bm1  = (const float*)d_in[9];
    const float* Wm2  = (const float*)d_in[10];
    const float* bm2  = (const float*)d_in[11];
    float* out   = (float*)d_out;
    float* fsums = (float*)d_ws;   // 2048*16*4 = 128 KB scratch

    fsums_kernel<<<NN / 16, 256, 0, stream>>>(x, W1, b1, W2, b2, fw, fsums);
    out_kernel<<<NN / 16, 256, 0, stream>>>(dist, norm, Wm1, bm1, Wm2, bm2, fsums, out);
}